// iVGAE_Decoder_57604101373964
// MI455X (gfx1250) — compile-verified
//
#include <hip/hip_runtime.h>
#include <math.h>

// ---------------------------------------------------------------------------
// 2-layer GCN on gfx1250 (MI455X).
//   deg/norm -> WMMA f32 GEMM -> self-loop init -> edge scatter (f32 atomics,
//   resolved in 192MB L2) -> bias+relu -> WMMA GEMM -> scatter into d_out ->
//   bias+sigmoid.
// Dense GEMMs use v_wmma_f32_16x16x4_f32 (fp32, matching the reference).
// All problem dims (64/128/64, chunk counts) are template constants so hot
// paths compile to shifts/masks + immediate-offset global loads.
// ---------------------------------------------------------------------------

typedef __attribute__((ext_vector_type(2))) float v2f;
typedef __attribute__((ext_vector_type(8))) float v8f;

// ---------------- degree / norm ----------------

__global__ void k_init_deg(float* __restrict__ deg, int n) {
  int i = blockIdx.x * blockDim.x + threadIdx.x;
  if (i < n) deg[i] = 1.0f;  // self-loop contribution
}

__global__ void k_accum_deg(const long long* __restrict__ ei, float* __restrict__ deg, int e) {
  int i = blockIdx.x * blockDim.x + threadIdx.x;
  if (i < e) {
    int d = (int)ei[(long long)e + i];  // edge_index[1][i]
    atomicAdd(deg + d, 1.0f);
  }
}

__global__ void k_rsqrt_inplace(float* __restrict__ deg, int n) {
  int i = blockIdx.x * blockDim.x + threadIdx.x;
  if (i < n) deg[i] = rsqrtf(deg[i]);  // deg >= 1 always (self-loop)
}

// ---------------- WMMA GEMM: C[M,FOUT] = A[M,K] @ W[K,FOUT], fp32 row-major.
// One wave computes one 16x16 output tile, stepping K by 4 with
// v_wmma_f32_16x16x4_f32.  Fragment layouts per CDNA5 ISA 7.12.2:
//   A 16x4 : lane l(0-15) row=l holds K={k0,k0+1}; lane l+16 holds K={k0+2,k0+3}
//   B 4x16 : VGPR0 lanes 0-15 = row k0 / lanes 16-31 = row k0+2; VGPR1 = k0+1 / k0+3
//   C/D    : VGPR i, lanes 0-15 -> M=i, lanes 16-31 -> M=i+8; N = lane&15
// K/FOUT are compile-time so the fully unrolled loop uses immediate offsets
// off two base pointers (A pair -> one global_load_b64 per step).
// ----------------

template <int K, int FOUT>
__global__ void k_gemm_wmma(const float* __restrict__ A, const float* __restrict__ W,
                            float* __restrict__ C, int M) {
  const int lane   = threadIdx.x & 31;
  const int waveId = threadIdx.x >> 5;
  const int tilesN = FOUT >> 4;
  const int tilesM = (M + 15) >> 4;
  const int tileId = blockIdx.x * (blockDim.x >> 5) + waveId;
  if (tileId >= tilesM * tilesN) return;  // wave-uniform: EXEC stays all-ones
  const int tm = tileId / tilesN;
  const int tn = tileId % tilesN;

  const int half = lane >> 4;     // selects K pair {0,1} vs {2,3}
  const int l15  = lane & 15;
  int mrow = tm * 16 + l15;
  if (mrow >= M) mrow = M - 1;    // clamp loads (stores predicated below)
  const int col = tn * 16 + l15;

  // Base pointers: all K-loop loads become immediate offsets from these.
  const float* __restrict__ arow = A + (size_t)mrow * K + half * 2;
  const float* __restrict__ wcol = W + (size_t)(half * 2) * FOUT + col;

  v8f acc = {};
#pragma unroll
  for (int k0 = 0; k0 < K; k0 += 4) {
    const v2f a = *(const v2f*)(arow + k0);        // 8B-aligned pair load
    v2f b;
    b.x = wcol[(size_t)k0 * FOUT];
    b.y = wcol[(size_t)(k0 + 1) * FOUT];
    // (neg_a, A, neg_b, B, c_mod, C, reuse_a, reuse_b)
    acc = __builtin_amdgcn_wmma_f32_16x16x4_f32(false, a, false, b, (short)0, acc,
                                                false, false);
  }

#pragma unroll
  for (int i = 0; i < 8; ++i) {
    const int row = tm * 16 + i + half * 8;
    if (row < M) C[(size_t)row * FOUT + col] = acc[i];
  }
}

// ---------------- self-loop init: agg[v] = xw[v] * dis[v]^2  (float4 lanes)
// Also serves as the full initialization of the accumulator buffer.

template <int CHUNKS>
__global__ void k_selfloop_init(const float4* __restrict__ xw4, const float* __restrict__ dis,
                                float4* __restrict__ agg4, int n) {
  const unsigned i = blockIdx.x * blockDim.x + threadIdx.x;
  if (i >= (unsigned)n * CHUNKS) return;
  const unsigned node = i / CHUNKS;  // CHUNKS is a power of two -> shift
  float w = dis[node];
  w *= w;
  float4 v = xw4[i];
  v.x *= w; v.y *= w; v.z *= w; v.w *= w;
  agg4[i] = v;
}

// ---------------- edge scatter: CHUNKS float4 lanes cover one edge's feature
// row; 32 consecutive lanes gather consecutive float4s of xw[src] (coalesced
// 512B), then 4 consecutive global_atomic_add_f32 into agg[dst] (L2-resident).

template <int CHUNKS>
__global__ void k_scatter(const long long* __restrict__ ei, const float* __restrict__ dis,
                          const float4* __restrict__ xw4, float* __restrict__ agg,
                          int e) {
  const unsigned tid = blockIdx.x * blockDim.x + threadIdx.x;
  if (tid >= (unsigned)e * CHUNKS) return;
  const unsigned edge  = tid / CHUNKS;   // power-of-two -> shift
  const unsigned chunk = tid % CHUNKS;   // power-of-two -> mask
  const int s = (int)ei[edge];                    // edge_index[0][edge]
  const int d = (int)ei[(unsigned)e + edge];      // edge_index[1][edge]
  const float nrm = dis[s] * dis[d];
  float4 m = xw4[(unsigned)s * CHUNKS + chunk];
  float* out = agg + ((size_t)d * CHUNKS + chunk) * 4;
  atomicAdd(out + 0, m.x * nrm);
  atomicAdd(out + 1, m.y * nrm);
  atomicAdd(out + 2, m.z * nrm);
  atomicAdd(out + 3, m.w * nrm);
}

// ---------------- epilogues ----------------

template <int F>
__global__ void k_bias_relu(float* __restrict__ h, const float* __restrict__ b, int n) {
  const unsigned i = blockIdx.x * blockDim.x + threadIdx.x;
  if (i >= (unsigned)n * F) return;
  const unsigned c = i % F;  // power-of-two -> mask
  h[i] = fmaxf(h[i] + b[c], 0.0f);
}

template <int F>
__global__ void k_bias_sigmoid(float* __restrict__ o, const float* __restrict__ b, int n) {
  const unsigned i = blockIdx.x * blockDim.x + threadIdx.x;
  if (i >= (unsigned)n * F) return;
  const unsigned c = i % F;
  const float v = o[i] + b[c];
  o[i] = 1.0f / (1.0f + __expf(-v));
}

// ---------------------------------------------------------------------------

static inline int cdiv_ll(long long a, int b) { return (int)((a + b - 1) / b); }

extern "C" void kernel_launch(void* const* d_in, const int* in_sizes, int n_in,
                              void* d_out, int out_size, void* d_ws, size_t ws_size,
                              hipStream_t stream) {
  const float*     x  = (const float*)d_in[0];       // [N, 64]
  const long long* ei = (const long long*)d_in[1];   // [2, E] int64
  const float*     W0 = (const float*)d_in[2];       // [64, 128]
  const float*     b0 = (const float*)d_in[3];       // [128]
  const float*     W1 = (const float*)d_in[4];       // [128, 64]
  const float*     b1 = (const float*)d_in[5];       // [64]

  constexpr int FIN  = 64;
  constexpr int FHID = 128;
  constexpr int FOUT = 64;
  const int n = in_sizes[0] / FIN;                   // 100000
  const int e = in_sizes[1] / 2;                     // 1600000

  // workspace layout (floats): dis[n] | xw[n*FHID] (reused for hw) | agg[n*FHID]
  float* dis = (float*)d_ws;
  float* xw  = dis + n;
  float* agg = xw + (size_t)n * FHID;

  const int BLK = 256;

  // ---- normalization ----
  k_init_deg<<<cdiv_ll(n, BLK), BLK, 0, stream>>>(dis, n);
  k_accum_deg<<<cdiv_ll(e, BLK), BLK, 0, stream>>>(ei, dis, e);
  k_rsqrt_inplace<<<cdiv_ll(n, BLK), BLK, 0, stream>>>(dis, n);

  // ---- layer 1: xw = X @ W0 ----
  {
    const long long tiles = (long long)((n + 15) / 16) * (FHID / 16);
    k_gemm_wmma<FIN, FHID><<<cdiv_ll(tiles, 4), 128, 0, stream>>>(x, W0, xw, n);
  }
  constexpr int CH1 = FHID / 4;  // float4 chunks per row (32)
  k_selfloop_init<CH1><<<cdiv_ll((long long)n * CH1, BLK), BLK, 0, stream>>>(
      (const float4*)xw, dis, (float4*)agg, n);
  k_scatter<CH1><<<cdiv_ll((long long)e * CH1, BLK), BLK, 0, stream>>>(
      ei, dis, (const float4*)xw, agg, e);
  k_bias_relu<FHID><<<cdiv_ll((long long)n * FHID, BLK), BLK, 0, stream>>>(
      agg, b0, n);  // agg now holds h

  // ---- layer 2: hw = h @ W1 (into reused xw region), agg2 lives in d_out ----
  {
    const long long tiles = (long long)((n + 15) / 16) * (FOUT / 16);
    k_gemm_wmma<FHID, FOUT><<<cdiv_ll(tiles, 4), 128, 0, stream>>>(agg, W1, xw, n);
  }
  float* out = (float*)d_out;
  constexpr int CH2 = FOUT / 4;  // 16
  k_selfloop_init<CH2><<<cdiv_ll((long long)n * CH2, BLK), BLK, 0, stream>>>(
      (const float4*)xw, dis, (float4*)out, n);
  k_scatter<CH2><<<cdiv_ll((long long)e * CH2, BLK), BLK, 0, stream>>>(
      ei, dis, (const float4*)xw, out, e);
  k_bias_sigmoid<FOUT><<<cdiv_ll((long long)n * FOUT, BLK), BLK, 0, stream>>>(
      out, b1, n);
}